// Model_50749333570303
// MI455X (gfx1250) — compile-verified
//
#include <hip/hip_runtime.h>
#include <hip/hip_bf16.h>
#include <math.h>

// ---------------------------------------------------------------------------
// Deformable-conv x2 + BN pipeline for MI455X (gfx1250, wave32).
// Fused: offset tensors (2 x 84MB naive) never materialized; offset convs run
// on V_WMMA_F32_16X16X4_F32 (fp32 matrix path — offsets feed bilinear coords).
// BN affine folded into stage-2 reads. Footprint ~25MB -> L2-resident.
// Round 2 change: branchless boundary gathers (clamp address + cndmask zero)
// instead of exec-mask-guarded loads, to unserialize the gather stream.
// ---------------------------------------------------------------------------

typedef __attribute__((ext_vector_type(2))) float v2f;
typedef __attribute__((ext_vector_type(8))) float v8f;

#define BATCH 8
#define H1 384
#define W1 384
#define CIN 3
#define HO 382
#define WO 382
#define HW1 (H1 * W1)
#define HW2 (HO * WO)              // 145924
#define NPIX (BATCH * HW2)         // 1167392  (divisible by 32 -> full-EXEC waves)
#define NGROUPS (NPIX / 32)        // 36481
#define NBLK 512
#define WPB 8                      // waves per block (256 threads, wave32)
#define TOTAL_WAVES (NBLK * WPB)

// workspace layout (floats)
#define WS_S 0                     // BN scale  s = gamma/sqrt(var+eps)
#define WS_T 1                     // BN shift  t = beta - mu*s
#define WS_PART 16                 // 2*NBLK partials (sum, sumsq)
#define WS_V2 2048                 // v2 buffer, NPIX floats

// Branchless zero-padded gather: unconditional load from clamped address,
// result zeroed by cndmask when out of bounds (torchvision clip+mask semantics).
__device__ __forceinline__ float zload(const float* __restrict__ img, int yi, int xi,
                                       int Hd, int Wd) {
    bool valid = ((unsigned)yi < (unsigned)Hd) & ((unsigned)xi < (unsigned)Wd);
    int yc = yi < 0 ? 0 : (yi > Hd - 1 ? Hd - 1 : yi);
    int xc = xi < 0 ? 0 : (xi > Wd - 1 ? Wd - 1 : xi);
    float v = img[yc * Wd + xc];
    return valid ? v : 0.0f;
}
// stage-2 read: v3 = s*v2 + t inside the 382x382 grid, literal 0 outside
__device__ __forceinline__ float zload_aff(const float* __restrict__ img, int yi, int xi,
                                           float aS, float aT) {
    bool valid = ((unsigned)yi < (unsigned)HO) & ((unsigned)xi < (unsigned)WO);
    int yc = yi < 0 ? 0 : (yi > HO - 1 ? HO - 1 : yi);
    int xc = xi < 0 ? 0 : (xi > WO - 1 ? WO - 1 : xi);
    float v = fmaf(aS, img[yc * WO + xc], aT);
    return valid ? v : 0.0f;
}

// ===========================================================================
// K1: offset-conv1 (WMMA f32 16x16x4, K=27 pad 28) + deform-conv1 -> v2,
//     plus per-block sum / sumsq partials for batch-norm.
// ===========================================================================
__global__ __launch_bounds__(256) void k1_deform1(
    const float* __restrict__ x1, const float* __restrict__ off1_w,
    const float* __restrict__ off1_b, const float* __restrict__ w1,
    const float* __restrict__ b1, float* __restrict__ wsf)
{
    __shared__ float s_off[WPB][32][19];   // per-wave private transpose buffer (pad 19)
    __shared__ float red[256];

    const int tid  = threadIdx.x;
    const int wv   = tid >> 5;
    const int lane = tid & 31;
    const int ln   = lane & 15;
    const int hi   = lane >> 4;
    float* v2buf = wsf + WS_V2;

    // A tiles: offset-conv weights [18][27], M padded to 32, K padded to 28.
    // A layout (16x4 f32): row M = lane&15, k = 4*ks + 2*(lane>>4) + vgpr.
    float a0[7][2], a1[7][2];
#pragma unroll
    for (int ks = 0; ks < 7; ++ks)
#pragma unroll
        for (int v = 0; v < 2; ++v) {
            int k = 4 * ks + 2 * hi + v;
            a0[ks][v] = (k < 27) ? off1_w[ln * 27 + k] : 0.0f;
            int m2 = 16 + ln;
            a1[ks][v] = (m2 < 18 && k < 27) ? off1_w[m2 * 27 + k] : 0.0f;
        }
    float w1r[27];
#pragma unroll
    for (int i = 0; i < 27; ++i) w1r[i] = w1[i];
    const float bias1 = b1[0];

    float lsum = 0.0f, lsq = 0.0f;
    const int wgid = blockIdx.x * WPB + wv;

    for (int g = wgid; g < NGROUPS; g += TOTAL_WAVES) {
        const int p0 = g * 32;

        // ---- offset conv: 2 N-tiles x 2 M-tiles x 7 K-steps of WMMA ----
#pragma unroll
        for (int nt = 0; nt < 2; ++nt) {
            int pix = p0 + nt * 16 + ln;
            int bb  = pix / HW2;  int rem = pix - bb * HW2;
            int oy  = rem / WO;   int ox  = rem - oy * WO;
            const float* xb = x1 + bb * (CIN * HW1);

            v8f acc0 = {0.f,0.f,0.f,0.f,0.f,0.f,0.f,0.f};
            v8f acc1 = {0.f,0.f,0.f,0.f,0.f,0.f,0.f,0.f};
#pragma unroll
            for (int ks = 0; ks < 7; ++ks) {
                v2f bv;
#pragma unroll
                for (int v = 0; v < 2; ++v) {
                    int r = 4 * ks + 2 * hi + v;     // patch index (c*9 + ky*3 + kx)
                    float val = 0.0f;
                    if (r < 27) {                    // r==27 only on pad step (uniform)
                        int c  = r / 9;  int tp = r - 9 * c;
                        int ky = tp / 3; int kx = tp - 3 * ky;
                        val = zload(xb + c * HW1, oy - 1 + 2 * ky, ox - 1 + 2 * kx, H1, W1);
                    }
                    bv[v] = val;
                }
                v2f av0 = { a0[ks][0], a0[ks][1] };
                v2f av1 = { a1[ks][0], a1[ks][1] };
                acc0 = __builtin_amdgcn_wmma_f32_16x16x4_f32(false, av0, false, bv,
                                                             (short)0, acc0, false, false);
                acc1 = __builtin_amdgcn_wmma_f32_16x16x4_f32(false, av1, false, bv,
                                                             (short)0, acc1, false, false);
            }
            // D layout: vgpr r -> M = r + 8*hi, N = lane&15. Transpose to LDS + bias.
#pragma unroll
            for (int r = 0; r < 8; ++r) {
                int m = r + 8 * hi;
                s_off[wv][nt * 16 + ln][m] = acc0[r] + off1_b[m];
                int mg = 16 + m;
                if (mg < 18) s_off[wv][nt * 16 + ln][mg] = acc1[r] + off1_b[mg];
            }
        }
        // cross-lane LDS RAW within the wave (per-wave region, no block barrier)
        asm volatile("s_wait_dscnt 0" ::: "memory");

        // ---- deformable bilinear sampling + 1-out-channel conv ----
        int p  = p0 + lane;
        int bb = p / HW2;  int rem = p - bb * HW2;
        int oy = rem / WO; int ox  = rem - oy * WO;
        const float* xb = x1 + bb * (CIN * HW1);
        float accv = 0.0f;
#pragma unroll
        for (int tp = 0; tp < 9; ++tp) {
            int ky = tp / 3, kx = tp - 3 * (tp / 3);
            float py = (float)(oy - 1 + 2 * ky) + s_off[wv][lane][2 * tp];
            float px = (float)(ox - 1 + 2 * kx) + s_off[wv][lane][2 * tp + 1];
            float fy = floorf(py), fx = floorf(px);
            int   iy = (int)fy,    ix = (int)fx;
            float dy = py - fy,    dx = px - fx;
            float w00 = (1.f - dy) * (1.f - dx), w01 = (1.f - dy) * dx;
            float w10 = dy * (1.f - dx),         w11 = dy * dx;
#pragma unroll
            for (int c = 0; c < CIN; ++c) {
                const float* im = xb + c * HW1;
                float sv = w00 * zload(im, iy,     ix,     H1, W1)
                         + w01 * zload(im, iy,     ix + 1, H1, W1)
                         + w10 * zload(im, iy + 1, ix,     H1, W1)
                         + w11 * zload(im, iy + 1, ix + 1, H1, W1);
                accv = fmaf(w1r[c * 9 + tp], sv, accv);
            }
        }
        float v2v = 2.0f * (accv + bias1);   // v2 = v1 + v1
        v2buf[p] = v2v;
        lsum += v2v;
        lsq  += v2v * v2v;
    }

    // ---- block reduction of BN partials ----
    red[tid] = lsum; __syncthreads();
    for (int s = 128; s > 0; s >>= 1) { if (tid < s) red[tid] += red[tid + s]; __syncthreads(); }
    if (tid == 0) wsf[WS_PART + 2 * blockIdx.x] = red[0];
    __syncthreads();
    red[tid] = lsq; __syncthreads();
    for (int s = 128; s > 0; s >>= 1) { if (tid < s) red[tid] += red[tid + s]; __syncthreads(); }
    if (tid == 0) wsf[WS_PART + 2 * blockIdx.x + 1] = red[0];
}

// ===========================================================================
// K2: finalize BN affine  s = gamma*rsqrt(var+eps), t = beta - mu*s
// ===========================================================================
__global__ __launch_bounds__(256) void k2_bnstats(const float* __restrict__ gamma,
                                                  const float* __restrict__ beta,
                                                  float* __restrict__ wsf)
{
    __shared__ float rs[256], rq[256];
    int tid = threadIdx.x;
    float s = 0.0f, q = 0.0f;
    for (int b = tid; b < NBLK; b += 256) {
        s += wsf[WS_PART + 2 * b];
        q += wsf[WS_PART + 2 * b + 1];
    }
    rs[tid] = s; rq[tid] = q; __syncthreads();
    for (int k = 128; k > 0; k >>= 1) {
        if (tid < k) { rs[tid] += rs[tid + k]; rq[tid] += rq[tid + k]; }
        __syncthreads();
    }
    if (tid == 0) {
        float mean = rs[0] / (float)NPIX;
        float var  = rq[0] / (float)NPIX - mean * mean;   // biased var (torch BN)
        float inv  = rsqrtf(var + 1e-5f);
        float sc   = gamma[0] * inv;
        wsf[WS_S]  = sc;
        wsf[WS_T]  = beta[0] - mean * sc;
    }
}

// ===========================================================================
// K4: offset-conv2 (WMMA, K=9 pad 12) + deform-conv2 on v3 = s*v2+t -> out
// ===========================================================================
__global__ __launch_bounds__(256) void k4_deform2(
    const float* __restrict__ off2_w, const float* __restrict__ off2_b,
    const float* __restrict__ w2, const float* __restrict__ b2,
    const float* __restrict__ wsf, float* __restrict__ out)
{
    __shared__ float s_off[WPB][32][19];
    const int tid  = threadIdx.x;
    const int wv   = tid >> 5;
    const int lane = tid & 31;
    const int ln   = lane & 15;
    const int hi   = lane >> 4;
    const float* v2buf = wsf + WS_V2;
    const float aS = wsf[WS_S];
    const float aT = wsf[WS_T];

    float a0[3][2], a1[3][2];
#pragma unroll
    for (int ks = 0; ks < 3; ++ks)
#pragma unroll
        for (int v = 0; v < 2; ++v) {
            int k = 4 * ks + 2 * hi + v;
            a0[ks][v] = (k < 9) ? off2_w[ln * 9 + k] : 0.0f;
            int m2 = 16 + ln;
            a1[ks][v] = (m2 < 18 && k < 9) ? off2_w[m2 * 9 + k] : 0.0f;
        }
    float w2r[9];
#pragma unroll
    for (int i = 0; i < 9; ++i) w2r[i] = w2[i];
    const float bias2 = b2[0];

    const int wgid = blockIdx.x * WPB + wv;
    for (int g = wgid; g < NGROUPS; g += TOTAL_WAVES) {
        const int p0 = g * 32;

#pragma unroll
        for (int nt = 0; nt < 2; ++nt) {
            int pix = p0 + nt * 16 + ln;
            int bb  = pix / HW2;  int rem = pix - bb * HW2;
            int oy  = rem / WO;   int ox  = rem - oy * WO;
            const float* vb = v2buf + bb * HW2;

            v8f acc0 = {0.f,0.f,0.f,0.f,0.f,0.f,0.f,0.f};
            v8f acc1 = {0.f,0.f,0.f,0.f,0.f,0.f,0.f,0.f};
#pragma unroll
            for (int ks = 0; ks < 3; ++ks) {
                v2f bv;
#pragma unroll
                for (int v = 0; v < 2; ++v) {
                    int r = 4 * ks + 2 * hi + v;
                    float val = 0.0f;
                    if (r < 9) {
                        int ky = r / 3; int kx = r - 3 * ky;
                        val = zload_aff(vb, oy - 1 + ky, ox - 1 + kx, aS, aT);
                    }
                    bv[v] = val;
                }
                v2f av0 = { a0[ks][0], a0[ks][1] };
                v2f av1 = { a1[ks][0], a1[ks][1] };
                acc0 = __builtin_amdgcn_wmma_f32_16x16x4_f32(false, av0, false, bv,
                                                             (short)0, acc0, false, false);
                acc1 = __builtin_amdgcn_wmma_f32_16x16x4_f32(false, av1, false, bv,
                                                             (short)0, acc1, false, false);
            }
#pragma unroll
            for (int r = 0; r < 8; ++r) {
                int m = r + 8 * hi;
                s_off[wv][nt * 16 + ln][m] = acc0[r] + off2_b[m];
                int mg = 16 + m;
                if (mg < 18) s_off[wv][nt * 16 + ln][mg] = acc1[r] + off2_b[mg];
            }
        }
        asm volatile("s_wait_dscnt 0" ::: "memory");

        int p  = p0 + lane;
        int bb = p / HW2;  int rem = p - bb * HW2;
        int oy = rem / WO; int ox  = rem - oy * WO;
        const float* vb = v2buf + bb * HW2;
        float accv = 0.0f;
#pragma unroll
        for (int tp = 0; tp < 9; ++tp) {
            int ky = tp / 3, kx = tp - 3 * (tp / 3);
            float py = (float)(oy - 1 + ky) + s_off[wv][lane][2 * tp];
            float px = (float)(ox - 1 + kx) + s_off[wv][lane][2 * tp + 1];
            float fy = floorf(py), fx = floorf(px);
            int   iy = (int)fy,    ix = (int)fx;
            float dy = py - fy,    dx = px - fx;
            float sv = (1.f - dy) * (1.f - dx) * zload_aff(vb, iy,     ix,     aS, aT)
                     + (1.f - dy) * dx         * zload_aff(vb, iy,     ix + 1, aS, aT)
                     + dy * (1.f - dx)         * zload_aff(vb, iy + 1, ix,     aS, aT)
                     + dy * dx                 * zload_aff(vb, iy + 1, ix + 1, aS, aT);
            accv = fmaf(w2r[tp], sv, accv);
        }
        out[p] = accv + bias2;
    }
}

// ===========================================================================
extern "C" void kernel_launch(void* const* d_in, const int* in_sizes, int n_in,
                              void* d_out, int out_size, void* d_ws, size_t ws_size,
                              hipStream_t stream) {
    const float* x1     = (const float*)d_in[0];
    const float* off1_w = (const float*)d_in[1];
    const float* off1_b = (const float*)d_in[2];
    const float* w1     = (const float*)d_in[3];
    const float* b1     = (const float*)d_in[4];
    const float* gamma  = (const float*)d_in[5];
    const float* beta   = (const float*)d_in[6];
    const float* off2_w = (const float*)d_in[7];
    const float* off2_b = (const float*)d_in[8];
    const float* w2     = (const float*)d_in[9];
    const float* b2     = (const float*)d_in[10];
    float* wsf = (float*)d_ws;
    float* out = (float*)d_out;

    hipLaunchKernelGGL(k1_deform1, dim3(NBLK), dim3(256), 0, stream,
                       x1, off1_w, off1_b, w1, b1, wsf);
    hipLaunchKernelGGL(k2_bnstats, dim3(1), dim3(256), 0, stream, gamma, beta, wsf);
    hipLaunchKernelGGL(k4_deform2, dim3(NBLK), dim3(256), 0, stream,
                       off2_w, off2_b, w2, b2, wsf, out);
}